// GCN_44710609551878
// MI455X (gfx1250) — compile-verified
//
#include <hip/hip_runtime.h>
#include <hip/hip_bf16.h>

// ---------------------------------------------------------------------------
// 2-layer GCN on MI455X (gfx1250).
//  - Dense GEMMs via V_WMMA_F32_16X16X4_F32 (exact f32, matrix pipe).
//  - Edge gather/scatter is the bandwidth cost (~2.5 GB) and is L2-resident
//    (all node-feature matrices total < 80 MB << 192 MB L2).
//  - Scatter-adds use native global_atomic_add_f32 at device scope via
//    inline asm (guaranteed single L2-side RMW, no CAS loop).
// ---------------------------------------------------------------------------

#define N_NODES 100000
#define N_EDGES 3200000
#define IN_F    16
#define HID     64
#define NCLS    32

typedef float v2f __attribute__((ext_vector_type(2)));
typedef float v8f __attribute__((ext_vector_type(8)));

// Native fp32 scatter-add, device scope (cross-WGP coherent, resolved in L2).
__device__ __forceinline__ void atomAddF(float* p, float v) {
    asm volatile("global_atomic_add_f32 %0, %1, off scope:SCOPE_DEV"
                 :: "v"(p), "v"(v) : "memory");
}

// ------------------------------ utility ------------------------------------

__global__ void gcn_zero_f32(float* __restrict__ p, int n) {
    int i = blockIdx.x * blockDim.x + threadIdx.x;
    if (i < n) p[i] = 0.0f;
}

__global__ void gcn_zero_f32x4(float4* __restrict__ p, int n4) {
    int i = blockIdx.x * blockDim.x + threadIdx.x;
    if (i < n4) p[i] = make_float4(0.f, 0.f, 0.f, 0.f);
}

// deg[col[e]] += 1
__global__ void gcn_degree(const int* __restrict__ ei, float* __restrict__ deg) {
    int e = blockIdx.x * blockDim.x + threadIdx.x;
    if (e < N_EDGES) atomAddF(&deg[ei[N_EDGES + e]], 1.0f);
}

// deg -> dinv in place
__global__ void gcn_dinv(float* __restrict__ deg) {
    int i = blockIdx.x * blockDim.x + threadIdx.x;
    if (i < N_NODES) {
        float d = deg[i];
        deg[i] = (d > 0.0f) ? __frsqrt_rn(d) : 0.0f;
    }
}

// norm[e] = dinv[row[e]] * dinv[col[e]]
__global__ void gcn_norm(const int* __restrict__ ei, const float* __restrict__ dinv,
                         float* __restrict__ norm) {
    int e = blockIdx.x * blockDim.x + threadIdx.x;
    if (e < N_EDGES) norm[e] = dinv[ei[e]] * dinv[ei[N_EDGES + e]];
}

// out[i] = b2[i % 32]  (segment_sum target starts at bias)
__global__ void gcn_init_out(float* __restrict__ out, const float* __restrict__ b2) {
    int i = blockIdx.x * blockDim.x + threadIdx.x;
    if (i < N_NODES * NCLS) out[i] = b2[i & (NCLS - 1)];
}

// ------------------------------ WMMA GEMMs ---------------------------------
// One wave owns one 16x16 output tile.
// A 16x4 f32 layout (ISA 7.12.2): lane<16 -> row=lane,    v0:K=0 v1:K=1
//                                 lane>=16 -> row=lane-16, v0:K=2 v1:K=3
// B 4x16: same K split across lane halves, lane&15 = column.
// C/D 16x16: VGPR v -> M = v + 8*(lane>>4), N = lane&15.

// h1[N,64] = x[N,16] @ W1[16,64]   block = (32,4): wave y owns columns 16*y
__global__ __launch_bounds__(128) void gcn_gemm1(const float* __restrict__ x,
                                                 const float* __restrict__ W1,
                                                 float* __restrict__ h1) {
    const int lane = threadIdx.x;
    const int n0   = threadIdx.y << 4;        // 0,16,32,48
    const int m0   = blockIdx.x << 4;
    const int mr   = lane & 15;
    const int kh   = (lane >> 4) << 1;        // 0 or 2
    v8f c = {};
#pragma unroll
    for (int kk = 0; kk < 4; ++kk) {
        const int kb = (kk << 2) + kh;        // K base for this lane half
        v2f a, b;
        a.x = x[(m0 + mr) * IN_F + kb];
        a.y = x[(m0 + mr) * IN_F + kb + 1];
        b.x = W1[kb * HID + n0 + mr];
        b.y = W1[(kb + 1) * HID + n0 + mr];
        c = __builtin_amdgcn_wmma_f32_16x16x4_f32(false, a, false, b,
                                                  (short)0, c, false, false);
    }
    const int rhi = (lane >> 4) << 3;         // 0 or 8
#pragma unroll
    for (int v = 0; v < 8; ++v)
        h1[(m0 + rhi + v) * HID + n0 + mr] = c[v];
}

// h2[N,32] = relu(agg1 + b1)[N,64] @ W2[64,32]   block = (32,2)
// Bias + ReLU fused into the A-operand load (saves a full 75 MB L2 pass).
__global__ __launch_bounds__(64) void gcn_gemm2(const float* __restrict__ agg,
                                                const float* __restrict__ b1,
                                                const float* __restrict__ W2,
                                                float* __restrict__ h2) {
    const int lane = threadIdx.x;
    const int n0   = threadIdx.y << 4;        // 0,16
    const int m0   = blockIdx.x << 4;
    const int mr   = lane & 15;
    const int kh   = (lane >> 4) << 1;
    v8f c = {};
#pragma unroll
    for (int kk = 0; kk < 16; ++kk) {
        const int kb = (kk << 2) + kh;
        v2f av, bv;
        av.x = fmaxf(agg[(m0 + mr) * HID + kb]     + b1[kb],     0.0f);
        av.y = fmaxf(agg[(m0 + mr) * HID + kb + 1] + b1[kb + 1], 0.0f);
        bv.x = W2[kb * NCLS + n0 + mr];
        bv.y = W2[(kb + 1) * NCLS + n0 + mr];
        c = __builtin_amdgcn_wmma_f32_16x16x4_f32(false, av, false, bv,
                                                  (short)0, c, false, false);
    }
    const int rhi = (lane >> 4) << 3;
#pragma unroll
    for (int v = 0; v < 8; ++v)
        h2[(m0 + rhi + v) * NCLS + n0 + mr] = c[v];
}

// ------------------------------ edge scatter -------------------------------
// Each thread handles 4 consecutive features of one edge (float4 gather,
// 4 native f32 atomic adds).  16 (resp. 8) consecutive lanes cover one
// edge's full feature row -> coalesced 256 B (128 B) gathers.
template <int FEAT>
__global__ void gcn_scatter(const int* __restrict__ ei, const float* __restrict__ norm,
                            const float* __restrict__ h, float* __restrict__ out) {
    const int CHUNKS = FEAT / 4;
    long long t = (long long)blockIdx.x * blockDim.x + threadIdx.x;
    if (t >= (long long)N_EDGES * CHUNKS) return;
    int e  = (int)(t / CHUNKS);
    int f4 = (int)(t % CHUNKS) * 4;
    int r  = ei[e];
    int cl = ei[N_EDGES + e];
    float nv = norm[e];
    const float4 hv = *reinterpret_cast<const float4*>(&h[(long long)r * FEAT + f4]);
    float* dst = &out[(long long)cl * FEAT + f4];
    atomAddF(dst + 0, hv.x * nv);
    atomAddF(dst + 1, hv.y * nv);
    atomAddF(dst + 2, hv.z * nv);
    atomAddF(dst + 3, hv.w * nv);
}

// ------------------------------ driver -------------------------------------

extern "C" void kernel_launch(void* const* d_in, const int* in_sizes, int n_in,
                              void* d_out, int out_size, void* d_ws, size_t ws_size,
                              hipStream_t stream) {
    const float* x  = (const float*)d_in[0];
    const int*   ei = (const int*)d_in[1];      // [2, E] int32
    const float* W1 = (const float*)d_in[2];
    const float* b1 = (const float*)d_in[3];
    const float* W2 = (const float*)d_in[4];
    const float* b2 = (const float*)d_in[5];
    float* out = (float*)d_out;

    // Workspace layout (floats): deg[N] | norm[E] | h1[N*64] | agg1[N*64]
    float* W    = (float*)d_ws;
    float* deg  = W;                                  // then dinv in place
    float* norm = W + N_NODES;
    float* h1   = norm + N_EDGES;                     // reused as h2 later
    float* agg1 = h1 + (long long)N_NODES * HID;

    const int T = 256;
    auto blk = [](long long n, int t) { return (unsigned)((n + t - 1) / t); };

    // normalization
    gcn_zero_f32<<<blk(N_NODES, T), T, 0, stream>>>(deg, N_NODES);
    gcn_zero_f32x4<<<blk((long long)N_NODES * HID / 4, T), T, 0, stream>>>(
        (float4*)agg1, N_NODES * HID / 4);
    gcn_degree<<<blk(N_EDGES, T), T, 0, stream>>>(ei, deg);
    gcn_dinv<<<blk(N_NODES, T), T, 0, stream>>>(deg);
    gcn_norm<<<blk(N_EDGES, T), T, 0, stream>>>(ei, deg, norm);

    // layer 1: GEMM (WMMA f32) -> scatter
    gcn_gemm1<<<N_NODES / 16, dim3(32, 4), 0, stream>>>(x, W1, h1);
    gcn_scatter<HID><<<blk((long long)N_EDGES * (HID / 4), T), T, 0, stream>>>(ei, norm, h1, agg1);

    // layer 2: GEMM (WMMA f32, bias+ReLU fused on input) -> init out -> scatter
    float* h2 = h1;  // h1 dead after scatter1
    gcn_gemm2<<<N_NODES / 16, dim3(32, 2), 0, stream>>>(agg1, b1, W2, h2);
    gcn_init_out<<<blk((long long)N_NODES * NCLS, T), T, 0, stream>>>(out, b2);
    gcn_scatter<NCLS><<<blk((long long)N_EDGES * (NCLS / 4), T), T, 0, stream>>>(ei, norm, h2, out);
}